// HarmonicOrderDetector_81449759802030
// MI455X (gfx1250) — compile-verified
//
#include <hip/hip_runtime.h>
#include <hip/hip_bf16.h>

typedef __attribute__((ext_vector_type(2))) float v2f;
typedef __attribute__((ext_vector_type(8))) float v8f;

#define HIDDEN 128
#define TOKENS_PER_BLOCK 128   // 8 waves * 16 tokens
#define NB 64
#define NP 1024

// ---------------------------------------------------------------------------
// Kernel 1: the 2->128->128->8 MLP.
// Layer 1 (fan-in 2) is VALU into LDS. Layers 2 and 3 use
// V_WMMA_F32_16X16X4_F32 (exact f32 math, matches the f32 reference).
// Each wave computes a 16-token x 128 tile (layer 2) then a 16x16 tile
// (layer 3, cols 8..15 padded with zeros and discarded).
// ---------------------------------------------------------------------------
__global__ __launch_bounds__(256) void mlp_wmma_kernel(
    const float* __restrict__ freqs, const float* __restrict__ mags,
    const float* __restrict__ W1, const float* __restrict__ b1,
    const float* __restrict__ W2, const float* __restrict__ b2,
    const float* __restrict__ W3, const float* __restrict__ b3,
    float* __restrict__ out)
{
  __shared__ float hbuf[TOKENS_PER_BLOCK * HIDDEN];   // 64 KB: holds H1, then H2

  const int tid       = threadIdx.x;
  const int blockTok0 = blockIdx.x * TOKENS_PER_BLOCK;

  // ---- Layer 1: H1 = relu([f,m] @ W1 + b1); 2 threads per token ----
  {
    const int token = tid >> 1;          // 0..127
    const int cbase = (tid & 1) * 64;    // columns 0..63 or 64..127
    const int g     = blockTok0 + token;
    const float f = freqs[g];
    const float m = mags[g];
    #pragma unroll 8
    for (int j = 0; j < 64; ++j) {
      const int c = cbase + j;
      float h = fmaf(f, W1[c], fmaf(m, W1[HIDDEN + c], b1[c]));
      hbuf[token * HIDDEN + c] = h > 0.0f ? h : 0.0f;
    }
  }
  __syncthreads();

  // wave32 fragment coordinates for V_WMMA_F32_16X16X4_F32
  const int lane  = tid & 31;
  const int wave  = tid >> 5;            // 0..7
  const int idx   = lane & 15;           // A row (M) / B col (N)
  const int half  = (lane >> 4) << 1;    // K sub-offset: lanes 0-15 -> 0, 16-31 -> 2
  const int wrow0 = wave * 16;           // this wave's first local token row

  // ---- Layer 2: H2 = relu(H1 @ W2 + b2), 8 N-tiles of 16 cols ----
  v8f acc[8];
  #pragma unroll
  for (int n = 0; n < 8; ++n) {
    const float bias = b2[n * 16 + idx];
    #pragma unroll
    for (int r = 0; r < 8; ++r) acc[n][r] = bias;
  }

  const float* aP = &hbuf[(wrow0 + idx) * HIDDEN + half];  // this lane's A row

  #pragma unroll 4
  for (int k = 0; k < HIDDEN; k += 4) {
    v2f a; a.x = aP[k]; a.y = aP[k + 1];
    #pragma unroll
    for (int n = 0; n < 8; ++n) {
      const float* bP = &W2[(k + half) * HIDDEN + n * 16 + idx];
      v2f b; b.x = bP[0]; b.y = bP[HIDDEN];
      acc[n] = __builtin_amdgcn_wmma_f32_16x16x4_f32(
          false, a, false, b, (short)0, acc[n], false, false);
    }
  }

  __syncthreads();   // all waves done reading H1

  // relu + write H2 back into this wave's own 16 rows of hbuf
  #pragma unroll
  for (int n = 0; n < 8; ++n) {
    #pragma unroll
    for (int r = 0; r < 8; ++r) {
      const int row = wrow0 + ((lane < 16) ? r : r + 8);
      const float v = acc[n][r];
      hbuf[row * HIDDEN + n * 16 + idx] = v > 0.0f ? v : 0.0f;
    }
  }
  __syncthreads();

  // ---- Layer 3: OUT = H2 @ W3 + b3 (8 real cols, tile padded to 16) ----
  v8f acc3;
  {
    const float bias = (idx < 8) ? b3[idx] : 0.0f;
    #pragma unroll
    for (int r = 0; r < 8; ++r) acc3[r] = bias;
  }
  #pragma unroll 4
  for (int k = 0; k < HIDDEN; k += 4) {
    v2f a; a.x = aP[k]; a.y = aP[k + 1];   // hbuf now holds H2
    v2f b;
    b.x = (idx < 8) ? W3[(k + half)     * 8 + idx] : 0.0f;
    b.y = (idx < 8) ? W3[(k + half + 1) * 8 + idx] : 0.0f;
    acc3 = __builtin_amdgcn_wmma_f32_16x16x4_f32(
        false, a, false, b, (short)0, acc3, false, false);
  }

  if (idx < 8) {
    const int gbase = blockTok0 + wrow0;
    #pragma unroll
    for (int r = 0; r < 8; ++r) {
      const int row = (lane < 16) ? r : r + 8;
      out[(gbase + row) * 8 + idx] = acc3[r];
    }
  }
}

// ---------------------------------------------------------------------------
// Kernel 2: fundamental frequency of the LAST batch row (reference returns
// fund[-1] only). One 1024-thread workgroup; thread i counts harmonic matches
// for peak i; thread 0 reproduces JAX's lexicographic argmax semantics.
// ---------------------------------------------------------------------------
__global__ __launch_bounds__(1024) void fundamental_kernel(
    const float* __restrict__ freqs, const float* __restrict__ mags,
    float* __restrict__ out)
{
  __shared__ float fr[NP];
  __shared__ float mg[NP];
  __shared__ int   hm[NP];

  const int i    = threadIdx.x;
  const int base = (NB - 1) * NP;
  fr[i] = freqs[base + i];
  mg[i] = mags[base + i];
  __syncthreads();

  const float fi = fr[i];
  int h = 0;
  for (int j = 0; j < NP; ++j) {
    if (j == i) continue;
    if (!(mg[j] > 0.0f)) continue;        // valid[j]
    const float fj = fr[j];
    bool match = false;
    #pragma unroll
    for (int order = 2; order <= 8; ++order)
      match = match || (fabsf(fj - fi * (float)order) < 2.0f);
    h += match ? 1 : 0;
  }
  const bool valid = mg[i] > 0.0f;
  hm[i] = (valid && h > 0) ? h : -1;      // h_masked
  __syncthreads();

  if (i == 0) {
    int max_h = -1;
    for (int t = 0; t < NP; ++t) max_h = (hm[t] > max_h) ? hm[t] : max_h;
    int best_idx = 0;
    float best_mag = -__builtin_inff();
    if (max_h > 0) {                       // any candidate exists
      for (int t = 0; t < NP; ++t)
        if (hm[t] == max_h && mg[t] > best_mag) { best_mag = mg[t]; best_idx = t; }
    } else {                               // fallback: first argmax of mags
      for (int t = 0; t < NP; ++t)
        if (mg[t] > best_mag) { best_mag = mg[t]; best_idx = t; }
    }
    out[NB * NP * 8] = fr[best_idx];
  }
}

extern "C" void kernel_launch(void* const* d_in, const int* in_sizes, int n_in,
                              void* d_out, int out_size, void* d_ws, size_t ws_size,
                              hipStream_t stream) {
  const float* freqs = (const float*)d_in[0];
  const float* mags  = (const float*)d_in[1];
  const float* W1    = (const float*)d_in[2];
  const float* b1    = (const float*)d_in[3];
  const float* W2    = (const float*)d_in[4];
  const float* b2    = (const float*)d_in[5];
  const float* W3    = (const float*)d_in[6];
  const float* b3    = (const float*)d_in[7];
  float* out = (float*)d_out;

  const int ntok   = in_sizes[0];               // 64 * 1024 = 65536
  const int blocks = ntok / TOKENS_PER_BLOCK;   // 512

  mlp_wmma_kernel<<<blocks, 256, 0, stream>>>(freqs, mags, W1, b1, W2, b2, W3, b3, out);
  fundamental_kernel<<<1, 1024, 0, stream>>>(freqs, mags, out);
  (void)d_ws; (void)ws_size; (void)n_in; (void)out_size;
}